// MissingValueImputer_29910152249390
// MI455X (gfx1250) — compile-verified
//
#include <hip/hip_runtime.h>
#include <math.h>

// ---------------------------------------------------------------------------
// MissingValueImputer for MI455X (gfx1250, wave32, WMMA)
// B=8, W=64 -> BW=512 rows; N=128 nodes; F=128 features; H=4 heads
// Attention stage is algebraically the identity (h is broadcast over nodes),
// so the kernel chain is:
//   adj   : normalize node_emb -> sim GEMM -> top-k select      (output #2)
//   C1    = x @ fp_W + fp_b                 [512x128, K=128]  (WMMA f32)
//   R     = concat_h relu(C1 @ gat_W[h]^T)  [512x512, K=128]  (WMMA f32)
//   out   = x*(1-mask) + (R @ op_W + op_b)*mask  [K=512]      (WMMA f32)
// ---------------------------------------------------------------------------

typedef float v2f __attribute__((ext_vector_type(2)));
typedef float v8f __attribute__((ext_vector_type(8)));

#define N_NODES 128
#define FDIM    128
#define NHEADS  4
#define BW      512

__device__ __forceinline__ v8f wmma4(v2f a, v2f b, v8f c) {
  // V_WMMA_F32_16X16X4_F32 : D = A(16x4) * B(4x16) + C(16x16), fp32 throughout
  return __builtin_amdgcn_wmma_f32_16x16x4_f32(
      /*neg_a=*/false, a, /*neg_b=*/false, b,
      /*c_mod=*/(short)0, c, /*reuse_a=*/false, /*reuse_b=*/false);
}

// --- adjacency ---------------------------------------------------------------

__global__ void normalize_rows(const float* __restrict__ emb,
                               float* __restrict__ nrm) {
  __shared__ float red[4];
  const int row = blockIdx.x, t = threadIdx.x;
  float v  = emb[row * FDIM + t];
  float ss = v * v;
#pragma unroll
  for (int off = 16; off > 0; off >>= 1) ss += __shfl_down(ss, off, 32);
  if ((t & 31) == 0) red[t >> 5] = ss;
  __syncthreads();
  const float total = red[0] + red[1] + red[2] + red[3];
  nrm[row * FDIM + t] = v / sqrtf(total);
}

__global__ void build_adj(const float* __restrict__ nrm,
                          float* __restrict__ adj,
                          const int* __restrict__ kptr) {
  __shared__ float rowv[FDIM];
  __shared__ float redv[N_NODES];
  __shared__ int   redi[N_NODES];
  const int row = blockIdx.x, t = threadIdx.x;
  rowv[t] = nrm[row * FDIM + t];
  __syncthreads();
  const float* cj = nrm + t * FDIM;
  float acc = 0.f;
#pragma unroll 8
  for (int f = 0; f < FDIM; ++f) acc = fmaf(rowv[f], cj[f], acc);

  // iterative top-k (argmax, ties -> lower index, matching jax.lax.top_k)
  float myval = acc;
  bool  sel   = false;
  const int kk = *kptr;
  for (int it = 0; it < kk; ++it) {
    redv[t] = myval;
    redi[t] = t;
    __syncthreads();
    for (int s = N_NODES / 2; s > 0; s >>= 1) {
      if (t < s) {
        const float v2 = redv[t + s];
        const int   i2 = redi[t + s];
        if (v2 > redv[t] || (v2 == redv[t] && i2 < redi[t])) {
          redv[t] = v2;
          redi[t] = i2;
        }
      }
      __syncthreads();
    }
    if (t == redi[0]) { sel = true; myval = -INFINITY; }
    __syncthreads();
  }
  adj[row * N_NODES + t] = sel ? 1.f : 0.f;
}

// --- GEMM 1: C1 = x @ fp_W + fp_b -------------------------------------------

__global__ void gemm_fp(const float* __restrict__ X, const float* __restrict__ Wm,
                        const float* __restrict__ bias, float* __restrict__ C) {
  const int lane = threadIdx.x & 31;
  const int wave = threadIdx.x >> 5;
  const int gw   = blockIdx.x * 8 + wave;   // 256 waves = 32x8 tiles
  const int tm   = gw >> 3;
  const int tn   = gw & 7;
  const int l16  = lane & 15, lh = lane >> 4;
  const int row0 = tm * 16, col0 = tn * 16;

  v8f c = {0.f, 0.f, 0.f, 0.f, 0.f, 0.f, 0.f, 0.f};
#pragma unroll 8
  for (int k0 = 0; k0 < FDIM; k0 += 4) {
    const int ka = k0 + 2 * lh;
    v2f a = *(const v2f*)(X + (row0 + l16) * FDIM + ka);
    v2f b;
    b.x = Wm[ka * FDIM + col0 + l16];
    b.y = Wm[(ka + 1) * FDIM + col0 + l16];
    c = wmma4(a, b, c);
  }
  const float bn = bias[col0 + l16];
#pragma unroll
  for (int r = 0; r < 8; ++r) {
    const int m = row0 + r + 8 * lh;
    C[m * FDIM + col0 + l16] = c[r] + bn;
  }
}

// --- GEMM 2: R[:, h*128+o] = relu(C1 @ gat_W[h]^T) --------------------------

__global__ void gemm_gat(const float* __restrict__ C1, const float* __restrict__ gatW,
                         float* __restrict__ R) {
  const int lane = threadIdx.x & 31;
  const int wave = threadIdx.x >> 5;
  const int gw   = blockIdx.x * 8 + wave;   // 1024 waves = 4 heads x 32x8 tiles
  const int h    = gw >> 8;
  const int rem  = gw & 255;
  const int tm   = rem >> 3;
  const int tn   = rem & 7;
  const int l16  = lane & 15, lh = lane >> 4;
  const int row0 = tm * 16, col0 = tn * 16;
  const float* Bh = gatW + h * FDIM * FDIM;   // [o][f], contract over f

  v8f c = {0.f, 0.f, 0.f, 0.f, 0.f, 0.f, 0.f, 0.f};
#pragma unroll 8
  for (int k0 = 0; k0 < FDIM; k0 += 4) {
    const int ka = k0 + 2 * lh;
    v2f a = *(const v2f*)(C1 + (row0 + l16) * FDIM + ka);
    v2f b = *(const v2f*)(Bh + (col0 + l16) * FDIM + ka);  // B[k][n]=gatW[h][n][k]
    c = wmma4(a, b, c);
  }
#pragma unroll
  for (int r = 0; r < 8; ++r) {
    const int m = row0 + r + 8 * lh;
    R[m * (NHEADS * FDIM) + h * FDIM + col0 + l16] = fmaxf(c[r], 0.f);
  }
}

// --- GEMM 3 + blend: out = x*(1-mask) + (R @ op_W + op_b)*mask --------------

__global__ void gemm_out(const float* __restrict__ R, const float* __restrict__ opW,
                         const float* __restrict__ opb, const float* __restrict__ X,
                         const float* __restrict__ Mk, float* __restrict__ out) {
  const int lane = threadIdx.x & 31;
  const int wave = threadIdx.x >> 5;
  const int gw   = blockIdx.x * 8 + wave;   // 256 waves = 32x8 tiles
  const int tm   = gw >> 3;
  const int tn   = gw & 7;
  const int l16  = lane & 15, lh = lane >> 4;
  const int row0 = tm * 16, col0 = tn * 16;
  const int K    = NHEADS * FDIM;           // 512

  v8f c = {0.f, 0.f, 0.f, 0.f, 0.f, 0.f, 0.f, 0.f};
#pragma unroll 8
  for (int k0 = 0; k0 < K; k0 += 4) {
    const int ka = k0 + 2 * lh;
    v2f a = *(const v2f*)(R + (row0 + l16) * K + ka);
    v2f b;
    b.x = opW[ka * FDIM + col0 + l16];
    b.y = opW[(ka + 1) * FDIM + col0 + l16];
    c = wmma4(a, b, c);
  }
  const float bn = opb[col0 + l16];
#pragma unroll
  for (int r = 0; r < 8; ++r) {
    const int m   = row0 + r + 8 * lh;
    const int idx = m * FDIM + col0 + l16;
    const float imp = c[r] + bn;
    const float mk  = Mk[idx];
    out[idx] = X[idx] * (1.f - mk) + imp * mk;
  }
}

// ---------------------------------------------------------------------------

extern "C" void kernel_launch(void* const* d_in, const int* in_sizes, int n_in,
                              void* d_out, int out_size, void* d_ws, size_t ws_size,
                              hipStream_t stream) {
  const float* x        = (const float*)d_in[0];   // (8,64,128)
  const float* mask     = (const float*)d_in[1];   // (8,64,128)
  const float* node_emb = (const float*)d_in[2];   // (128,128)
  const float* gat_W    = (const float*)d_in[3];   // (4,128,128)
  // d_in[4] = gat_a : algebraically eliminated (attention is identity here)
  const float* fp_W     = (const float*)d_in[5];   // (128,128)
  const float* fp_b     = (const float*)d_in[6];   // (128,)
  const float* op_W     = (const float*)d_in[7];   // (512,128)
  const float* op_b     = (const float*)d_in[8];   // (128,)
  const int*   kptr     = (const int*)d_in[9];     // scalar k

  float* out = (float*)d_out;                      // (8,64,128) = 65536
  float* adj = out + BW * N_NODES;                 // (128,128)  = 16384

  float* ws   = (float*)d_ws;
  float* nrm  = ws;                                // 16384 floats
  float* C1   = ws + 16384;                        // 65536 floats
  float* R    = ws + 16384 + 65536;                // 262144 floats

  normalize_rows<<<N_NODES, FDIM, 0, stream>>>(node_emb, nrm);
  build_adj     <<<N_NODES, N_NODES, 0, stream>>>(nrm, adj, kptr);
  gemm_fp       <<<32, 256, 0, stream>>>(x, fp_W, fp_b, C1);
  gemm_gat      <<<128, 256, 0, stream>>>(C1, gat_W, R);
  gemm_out      <<<32, 256, 0, stream>>>(R, op_W, op_b, x, mask, out);
}